// BioRNN_60138132079085
// MI455X (gfx1250) — compile-verified
//
#include <hip/hip_runtime.h>
#include <math.h>

typedef float v2f __attribute__((ext_vector_type(2)));
typedef float v4f __attribute__((ext_vector_type(4)));
typedef float v8f __attribute__((ext_vector_type(8)));

#define N_TOT   3840
#define N_INP   128
#define KDIM    3968          // N_INP + N_TOT (fused [x | h] GEMM)
#define KB8     496           // KDIM / 8  (two 16x16x4 WMMA k-blocks per group)
#define BATCH   64
#define TSTEPS  200
#define N_PFCD  1024
#define DECAYC  0.2f          // DT/TAU = 10/50
#define NNOISE  0.01f

// region starts (within the N_TOT=3840 neuron vector)
#define SR_ES0   0
#define SR_ED0   512
#define SR_IN0   1536
#define PFC_ES0  1920
#define PFC_ED0  2432
#define PFC_IN0  3456

#define TILE_FLOATS (KB8 * 32 * 4)   // 63488 floats per 16-wide tile strip

// Fragment-order index for element (tile, l15, k):
//   k -> kblk=k>>2 (16x16x4 block), khalf=(k>>1)&1 selects lane half,
//   two kblks packed per 16B lane chunk: j = ((kblk&1)<<1) | (k&1)
__device__ __forceinline__ size_t frag_idx(int tile, int l15, int k) {
    int kblk = k >> 2;
    int lane = (((k >> 1) & 1) << 4) | l15;
    int j    = ((kblk & 1) << 1) | (k & 1);
    return ((size_t)(tile * KB8 + (kblk >> 1)) * 32 + lane) * 4 + j;
}

// ---------------------------------------------------------------------------
// One-time prep: Wsw = swizzled [ w_in (128 x 3840) ; |w_rec|*mask (3840x3840) ]
// ---------------------------------------------------------------------------
__global__ void biornn_prep_wc(const float* __restrict__ w_in,
                               const float* __restrict__ w_rec,
                               const float* __restrict__ mask,
                               float* __restrict__ Wsw) {
    long i = (long)blockIdx.x * blockDim.x + threadIdx.x;
    long tot = (long)KDIM * N_TOT;
    if (i >= tot) return;
    int k = (int)(i / N_TOT);          // 0..3967  (row of Wc)
    int n = (int)(i % N_TOT);          // 0..3839  (col)
    float v;
    if (k < N_INP) {
        v = w_in[(long)k * N_TOT + n];
    } else {
        long e = (long)(k - N_INP) * N_TOT + n;
        v = fabsf(w_rec[e]) * mask[e];
    }
    Wsw[frag_idx(n >> 4, n & 15, k)] = v;
}

// ---------------------------------------------------------------------------
// One-time init: Asw[b] = swizzled [ x[b,0,:] , h0[b,:] ] ; ime buffer = i_me0
// ---------------------------------------------------------------------------
__global__ void biornn_init_act(const float* __restrict__ x,
                                const float* __restrict__ h0,
                                float* __restrict__ Asw) {
    int i = blockIdx.x * blockDim.x + threadIdx.x;
    if (i >= BATCH * KDIM) return;
    int m = i / KDIM, k = i % KDIM;
    float v;
    if (k < N_INP) v = x[(long)m * TSTEPS * N_INP + k];       // t = 0 slice
    else           v = h0[(long)m * N_TOT + (k - N_INP)];
    Asw[frag_idx(m >> 4, m & 15, k)] = v;
}

__global__ void biornn_init_ime(const float* __restrict__ i_me0,
                                float* __restrict__ ime) {
    int i = blockIdx.x * blockDim.x + threadIdx.x;
    if (i >= BATCH * N_PFCD) return;
    ime[i] = i_me0[i];
}

// ---------------------------------------------------------------------------
// Per-step GEMM: pre[64,3840] = A[64,3968] @ Wc[3968,3840] + b + 0.01*noise_t
// fp32 WMMA 16x16x4, fragment-order operands: 1 x b128 load per WMMA, fully
// coalesced. Grid: 240 blocks (n-tiles) x 128 threads (4 waves = 4 m-tiles).
// ---------------------------------------------------------------------------
__global__ void biornn_gemm_pre(const float* __restrict__ Asw,
                                const float* __restrict__ Wsw,
                                const float* __restrict__ bias,
                                const float* __restrict__ noise_t,
                                float* __restrict__ pre) {
    const int lane  = threadIdx.x & 31;
    const int mtile = threadIdx.x >> 5;           // 0..3   (M = 64)
    const int ntile = blockIdx.x;                 // 0..239 (N = 3840)
    const int l15   = lane & 15;
    const int khalf = lane >> 4;                  // 0 or 1

    const v4f* __restrict__ afrag =
        (const v4f*)(Asw + ((size_t)mtile * KB8 * 32 + lane) * 4);
    const v4f* __restrict__ bfrag =
        (const v4f*)(Wsw + ((size_t)ntile * KB8 * 32 + lane) * 4);

    v8f c = {};
#pragma unroll 4
    for (int kb = 0; kb < KB8; ++kb) {
        v4f av = afrag[(size_t)kb * 32];          // 16B per lane, 512B per wave
        v4f bv = bfrag[(size_t)kb * 32];
        v2f a0 = {av[0], av[1]};
        v2f b0 = {bv[0], bv[1]};
        c = __builtin_amdgcn_wmma_f32_16x16x4_f32(false, a0, false, b0,
                                                  (short)0, c, false, false);
        v2f a1 = {av[2], av[3]};
        v2f b1 = {bv[2], bv[3]};
        c = __builtin_amdgcn_wmma_f32_16x16x4_f32(false, a1, false, b1,
                                                  (short)0, c, false, false);
    }

    // C/D layout: VGPR r -> row r (lanes 0-15) / row r+8 (lanes 16-31)
    const int  ncol = ntile * 16 + l15;
    const float bcst = bias[ncol];
#pragma unroll
    for (int r = 0; r < 8; ++r) {
        int row = mtile * 16 + r + khalf * 8;
        long idx = (long)row * N_TOT + ncol;
        pre[idx] = c[r] + bcst + NNOISE * noise_t[idx];
    }
}

// ---------------------------------------------------------------------------
// Per-step pointwise update: dendrite/soma nonlinearity, h decay, i_me decay.
// Writes h_new into the swizzled activation buffer (and a linear copy of the
// SR-soma block for the output projection). ime is double-buffered.
// Also stages x[:, t+1, :] into Asw for the next step.
// ---------------------------------------------------------------------------
__global__ void biornn_update(const float* __restrict__ pre,
                              float* __restrict__ Asw,
                              float* __restrict__ hsr,
                              const float* __restrict__ ime_old,
                              float* __restrict__ ime_new,
                              const float* __restrict__ alpha_me,
                              const float* __restrict__ dend2soma,
                              const float* __restrict__ x,
                              int t) {
    int i = blockIdx.x * blockDim.x + threadIdx.x;
    if (i >= BATCH * N_TOT) return;
    int b = i / N_TOT, n = i % N_TOT;
    const float* prow = pre + (long)b * N_TOT;
    float p   = prow[n];
    float d2s = dend2soma[0];
    float r;
    if (n < SR_ED0) {                       // SR soma
        float dend = tanhf(prow[SR_ED0 + n]) + tanhf(prow[SR_ED0 + 512 + n]);
        r = fmaxf(0.f, p + d2s * dend);
    } else if (n < SR_IN0) {                // SR dendrite
        r = tanhf(p);
    } else if (n < PFC_ES0) {               // SR inhibitory
        r = fmaxf(0.f, p);
    } else if (n < PFC_ED0) {               // PFC soma (recompute i_me_new from old)
        int j = n - PFC_ES0;
        float dend = 0.f;
#pragma unroll
        for (int br = 0; br < 2; ++br) {
            int   di  = j + br * 512;
            float pd  = prow[PFC_ED0 + di];
            float a   = alpha_me[di];
            float imn = (1.f - a) * ime_old[(long)b * N_PFCD + di] + a * pd;
            dend += tanhf(pd + imn);
        }
        r = fmaxf(0.f, p + d2s * dend);
    } else if (n < PFC_IN0) {               // PFC dendrite: update + write i_me
        int   di  = n - PFC_ED0;
        float a   = alpha_me[di];
        float imn = (1.f - a) * ime_old[(long)b * N_PFCD + di] + a * p;
        ime_new[(long)b * N_PFCD + di] = imn;
        r = tanhf(p + imn);
    } else {                                // PFC inhibitory
        r = fmaxf(0.f, p);
    }
    size_t hi = frag_idx(b >> 4, b & 15, N_INP + n);
    float hnew = (1.f - DECAYC) * Asw[hi] + DECAYC * r;
    Asw[hi] = hnew;
    if (n < SR_ED0) hsr[(long)b * 512 + n] = hnew;   // linear copy for out_proj

    // stage next timestep's input slice (swizzled)
    if (n < N_INP && (t + 1) < TSTEPS) {
        Asw[frag_idx(b >> 4, b & 15, n)] =
            x[((long)b * TSTEPS + (t + 1)) * N_INP + n];
    }
}

// ---------------------------------------------------------------------------
// Per-step output projection: y[b,t,:] = h_new[b, 0:512] @ w_out (512x4)
// ---------------------------------------------------------------------------
__global__ void biornn_out(const float* __restrict__ hsr,
                           const float* __restrict__ w_out,
                           float* __restrict__ out,
                           int t) {
    int i = threadIdx.x;                    // 0..255
    int b = i >> 2, o = i & 3;
    const float* h = hsr + (long)b * 512;
    float acc = 0.f;
    for (int j = 0; j < 512; ++j)
        acc = fmaf(h[j], w_out[j * 4 + o], acc);
    out[((long)b * TSTEPS + t) * 4 + o] = acc;   // output is (B, T, 4)
}

// ---------------------------------------------------------------------------
extern "C" void kernel_launch(void* const* d_in, const int* in_sizes, int n_in,
                              void* d_out, int out_size, void* d_ws, size_t ws_size,
                              hipStream_t stream) {
    const float* x      = (const float*)d_in[0];
    const float* h0     = (const float*)d_in[1];
    const float* i_me0  = (const float*)d_in[2];
    const float* noise  = (const float*)d_in[3];
    const float* w_rec  = (const float*)d_in[4];
    const float* w_in   = (const float*)d_in[5];
    const float* bias   = (const float*)d_in[6];
    const float* w_out  = (const float*)d_in[7];
    const float* mask   = (const float*)d_in[8];
    const float* alpha  = (const float*)d_in[9];
    const float* d2s    = (const float*)d_in[10];

    // workspace (floats): Wsw | Asw | pre | imeA | imeB | hsr  (~63.6 MB)
    float* ws   = (float*)d_ws;
    size_t off  = 0;
    float* Wsw  = ws + off; off += (size_t)(N_TOT / 16) * TILE_FLOATS;
    float* Asw  = ws + off; off += (size_t)(BATCH / 16) * TILE_FLOATS;
    float* pre  = ws + off; off += (size_t)BATCH * N_TOT;
    float* imeA = ws + off; off += (size_t)BATCH * N_PFCD;
    float* imeB = ws + off; off += (size_t)BATCH * N_PFCD;
    float* hsr  = ws + off;

    {
        long tot = (long)KDIM * N_TOT;
        biornn_prep_wc<<<(unsigned)((tot + 255) / 256), 256, 0, stream>>>(
            w_in, w_rec, mask, Wsw);
    }
    biornn_init_act<<<(BATCH * KDIM + 255) / 256, 256, 0, stream>>>(x, h0, Asw);
    biornn_init_ime<<<(BATCH * N_PFCD + 255) / 256, 256, 0, stream>>>(i_me0, imeA);

    for (int t = 0; t < TSTEPS; ++t) {
        const float* noise_t = noise + (size_t)t * BATCH * N_TOT;
        biornn_gemm_pre<<<N_TOT / 16, 128, 0, stream>>>(Asw, Wsw, bias, noise_t, pre);
        const float* iold = (t & 1) ? imeB : imeA;
        float*       inew = (t & 1) ? imeA : imeB;
        biornn_update<<<(BATCH * N_TOT + 255) / 256, 256, 0, stream>>>(
            pre, Asw, hsr, iold, inew, alpha, d2s, x, t);
        biornn_out<<<1, 256, 0, stream>>>(hsr, w_out, (float*)d_out, t);
    }
}